// Quantization4bit_24558622998914
// MI455X (gfx1250) — compile-verified
//
#include <hip/hip_runtime.h>
#include <hip/hip_fp16.h>
#include <cstdint>

// ---------------------------------------------------------------------------
// Quantization4bit for MI455X (gfx1250, wave32).
//   pass1: global min/max reduction (512 blocks, b128 loads, shfl_xor wave32)
//   pass2: finalize scale/bias + codebook midpoints (1 tiny block)
//   pass3: normalize + sorted-codebook nearest lookup, fed by the Tensor Data
//          Mover (tensor_load_to_lds, TENSORcnt) with LDS double buffering,
//          NT b128 stores so the output stream doesn't evict the L2-resident
//          input (32 MiB << 192 MB L2).
// Memory-bound floor: 32MB HBM read + 32MB HBM write ~ 3us @ 23.3 TB/s.
// q4_transform is defined FIRST so the disasm snippet shows the TDM path.
// ---------------------------------------------------------------------------

#define THREADS      256
#define RED_BLOCKS   512
#define TILE_FLOATS  4096            // 16 KiB tile
#define TILE_BYTES   (TILE_FLOATS * 4)
#define XFORM_BLOCKS 512
#define TILES_PER_BLOCK 4            // 512*4*4096 = 8,388,608 = 2048*4096

typedef unsigned int v4u __attribute__((ext_vector_type(4)));
typedef int          v8i __attribute__((ext_vector_type(8)));
typedef int          v4i __attribute__((ext_vector_type(4)));
typedef float        v4f __attribute__((ext_vector_type(4)));  // native vec4

// --- TDM: issue a 1-D tile DMA (global -> LDS), data_size = 4 bytes. -------
__device__ __forceinline__ void tdm_load_1d(uint32_t lds_off, uint64_t gaddr,
                                            int nelem) {
  v4u g0;
  g0.x = 1u;                                          // count=1, user mode
  g0.y = lds_off;                                     // LDS byte address
  g0.z = (uint32_t)(gaddr & 0xFFFFFFFFull);           // global_addr[31:0]
  g0.w = (uint32_t)((gaddr >> 32) & 0x01FFFFFFull)    // global_addr[56:32]
         | (2u << 30);                                // type = 2 ("image")
  v8i g1;
  g1[0] = (int)(2u << 16);                            // data_size = 2 (4B)
  g1[1] = (int)(((uint32_t)nelem & 0xFFFFu) << 16);   // tensor_dim0[15:0]
  g1[2] = (int)(((uint32_t)nelem >> 16) | (1u << 16));// dim0 hi, tensor_dim1=1
  g1[3] = (int)((uint32_t)nelem << 16);               // tile_dim0
  g1[4] = 0;                                          // tile_dim1/2 unused
  g1[5] = nelem;                                      // tensor_dim0_stride
  g1[6] = 0;
  g1[7] = 0;
  v4i z4 = {0, 0, 0, 0};
#if __clang_major__ >= 23
  v8i z8 = {0, 0, 0, 0, 0, 0, 0, 0};
  __builtin_amdgcn_tensor_load_to_lds(g0, g1, z4, z4, z8, 0);
#else
  __builtin_amdgcn_tensor_load_to_lds(g0, g1, z4, z4, 0);
#endif
}

// --- sorted-codebook nearest lookup: count 14 midpoints (branchless) -------
__device__ __forceinline__ float cb_lookup(float v, const float* c,
                                           const float* m) {
  float r = c[0];
#pragma unroll
  for (int j = 0; j < 14; ++j) r = (v > m[j]) ? c[j + 1] : r;
  return r;
}

// ---------------------------------------------------------------------------
// Kernel 3 (defined first for disasm visibility): TDM-pipelined transform.
// Wave 0 drives the Tensor Data Mover (double-buffered LDS tiles); all 8
// waves compute out of LDS (conflict-free ds_load_b128) and store NT b128.
// ---------------------------------------------------------------------------
__global__ void __launch_bounds__(THREADS)
q4_transform(const float* __restrict__ x,
             float* __restrict__ out,
             const float* __restrict__ params) {
  __shared__ float buf[2][TILE_FLOATS];  // 32 KiB of the 320 KiB WGP LDS
  const int tid = threadIdx.x;

  const float scale = params[0];
  const float bias  = params[1];
  float c[15], m[14];
#pragma unroll
  for (int i = 0; i < 15; ++i) c[i] = params[2 + i];
#pragma unroll
  for (int j = 0; j < 14; ++j) m[j] = params[17 + j];

  const int firstTile = blockIdx.x * TILES_PER_BLOCK;
  const uint64_t xbase = (uint64_t)(uintptr_t)x;

  if (tid < 32) {  // wave 0 is the DMA driver
    tdm_load_1d((uint32_t)(uintptr_t)&buf[0][0],
                xbase + (uint64_t)firstTile * TILE_BYTES, TILE_FLOATS);
  }

  for (int t = 0; t < TILES_PER_BLOCK; ++t) {
    if (tid < 32) {
      if (t + 1 < TILES_PER_BLOCK) {
        // issue tile t+1 into the buffer freed at end of iteration t-1,
        // then wait until only that DMA is in flight => tile t has landed
        // (TDM ops complete in order within a wave).
        tdm_load_1d((uint32_t)(uintptr_t)&buf[(t + 1) & 1][0],
                    xbase + (uint64_t)(firstTile + t + 1) * TILE_BYTES,
                    TILE_FLOATS);
        __builtin_amdgcn_s_wait_tensorcnt(1);
      } else {
        __builtin_amdgcn_s_wait_tensorcnt(0);
      }
    }
    __syncthreads();  // publish tile t to all waves

    const v4f* __restrict__ src = (const v4f*)&buf[t & 1][0];
    v4f* dst = (v4f*)out + (size_t)(firstTile + t) * (TILE_FLOATS / 4);
#pragma unroll
    for (int k = 0; k < 4; ++k) {
      const v4f v = src[tid + k * THREADS];  // lane L hits banks 4L..4L+3
      v4f r;
      r.x = cb_lookup(fmaf(v.x, scale, bias), c, m);
      r.y = cb_lookup(fmaf(v.y, scale, bias), c, m);
      r.z = cb_lookup(fmaf(v.z, scale, bias), c, m);
      r.w = cb_lookup(fmaf(v.w, scale, bias), c, m);
      __builtin_nontemporal_store(r, &dst[tid + k * THREADS]);
    }
    __syncthreads();  // all waves done reading buf[t&1] before DMA reuses it
  }
}

// ---------------------------------------------------------------------------
// Kernel 1: per-block min/max partials -> ws[0..1023]
// ---------------------------------------------------------------------------
__global__ void __launch_bounds__(THREADS)
q4_minmax_partial(const float4* __restrict__ x4,
                  float* __restrict__ partials) {
  const int gid = blockIdx.x * THREADS + threadIdx.x;
  const int stride = RED_BLOCKS * THREADS;   // 131072 float4s
  float lo = INFINITY, hi = -INFINITY;
#pragma unroll
  for (int i = 0; i < 16; ++i) {
    float4 v = x4[gid + i * stride];
    lo = fminf(lo, fminf(fminf(v.x, v.y), fminf(v.z, v.w)));
    hi = fmaxf(hi, fmaxf(fmaxf(v.x, v.y), fmaxf(v.z, v.w)));
  }
#pragma unroll
  for (int off = 16; off > 0; off >>= 1) {   // wave32 butterfly
    lo = fminf(lo, __shfl_xor(lo, off, 32));
    hi = fmaxf(hi, __shfl_xor(hi, off, 32));
  }
  __shared__ float slo[8], shi[8];
  const int wave = threadIdx.x >> 5, lane = threadIdx.x & 31;
  if (lane == 0) { slo[wave] = lo; shi[wave] = hi; }
  __syncthreads();
  if (threadIdx.x == 0) {
    float l = slo[0], h = shi[0];
#pragma unroll
    for (int w = 1; w < 8; ++w) { l = fminf(l, slo[w]); h = fmaxf(h, shi[w]); }
    partials[2 * blockIdx.x]     = l;
    partials[2 * blockIdx.x + 1] = h;
  }
}

// ---------------------------------------------------------------------------
// Kernel 2: fold 512 partial pairs; emit params = [scale, bias, c[15], m[14]]
// ---------------------------------------------------------------------------
__global__ void __launch_bounds__(THREADS)
q4_finalize(const float* __restrict__ partials,
            const __half* __restrict__ codebook,
            float* __restrict__ params) {
  const int tid = threadIdx.x;  // 256 threads
  float lo = fminf(partials[2 * tid],     partials[2 * (tid + 256)]);
  float hi = fmaxf(partials[2 * tid + 1], partials[2 * (tid + 256) + 1]);
#pragma unroll
  for (int off = 16; off > 0; off >>= 1) {
    lo = fminf(lo, __shfl_xor(lo, off, 32));
    hi = fmaxf(hi, __shfl_xor(hi, off, 32));
  }
  __shared__ float slo[8], shi[8];
  const int wave = tid >> 5, lane = tid & 31;
  if (lane == 0) { slo[wave] = lo; shi[wave] = hi; }
  __syncthreads();
  if (tid == 0) {
    float l = slo[0], h = shi[0];
#pragma unroll
    for (int w = 1; w < 8; ++w) { l = fminf(l, slo[w]); h = fmaxf(h, shi[w]); }
    // x_norm = (x - l)/(h - l)*2 - 1  ==  x*scale + bias
    const float scale = 2.0f / (h - l);
    const float bias  = -l * scale - 1.0f;
    params[0] = scale;
    params[1] = bias;
    float c[15];
#pragma unroll
    for (int i = 0; i < 15; ++i) {
      c[i] = __half2float(codebook[i]);
      params[2 + i] = c[i];
    }
#pragma unroll
    for (int j = 0; j < 14; ++j)
      params[17 + j] = 0.5f * (c[j] + c[j + 1]);
  }
}

// ---------------------------------------------------------------------------
extern "C" void kernel_launch(void* const* d_in, const int* in_sizes, int n_in,
                              void* d_out, int out_size, void* d_ws,
                              size_t ws_size, hipStream_t stream) {
  const float*  x  = (const float*)d_in[0];        // [2048,4096] fp32
  const __half* cb = (const __half*)d_in[1];       // [15] fp16 codebook
  float*        out = (float*)d_out;
  float*        ws  = (float*)d_ws;
  float* partials = ws;          // 1024 floats (512 blocks x {min,max})
  float* params   = ws + 1024;   // 31 floats: scale, bias, c[15], m[14]

  q4_minmax_partial<<<RED_BLOCKS, THREADS, 0, stream>>>((const float4*)x,
                                                        partials);
  q4_finalize<<<1, THREADS, 0, stream>>>(partials, cb, params);
  q4_transform<<<XFORM_BLOCKS, THREADS, 0, stream>>>(x, out, params);
}